// SSMKernel_8306466750954
// MI455X (gfx1250) — compile-verified
//
#include <hip/hip_runtime.h>

typedef __attribute__((ext_vector_type(2))) float v2f;
typedef __attribute__((ext_vector_type(8))) float v8f;

#define BATCH 8
#define SEQ   8192
#define DN    128
#define CH    128              // chunk length (timesteps per block)
#define NCH   (SEQ / CH)       // 64 chunks

// ---------------------------------------------------------------------------
// Pass 1: per-chunk local scan (h0 = 0) and local output y_local[t] = C[t]·h_local[t].
// One lane per state dim n (128 threads = 4 wave32). B/C reads are coalesced
// (lane index is the fastest-varying dim). dt/x staged in LDS. Per-t dot via
// wave32 shfl_xor reduction, cross-wave combine via a tiny LDS tile.
// Also emits chunk summaries: aprod[n] = exp(A[n]*sum(dt)), hloc[n] = h_local[end].
// ---------------------------------------------------------------------------
__global__ void ssm_pass1(const float* __restrict__ x, const float* __restrict__ dt,
                          const float* __restrict__ B, const float* __restrict__ C,
                          const float* __restrict__ A,
                          float* __restrict__ y, float* __restrict__ aprod,
                          float* __restrict__ hloc)
{
    __shared__ float dt_s[CH];
    __shared__ float dtx_s[CH];
    __shared__ float ypart[CH * 4];

    const int k = blockIdx.x, b = blockIdx.y;
    const int tid = threadIdx.x;
    const int lane = tid & 31, wv = tid >> 5;
    const size_t sbase = (size_t)b * SEQ + (size_t)k * CH;

    float d = dt[sbase + tid];
    dt_s[tid]  = d;
    dtx_s[tid] = d * x[sbase + tid];

    const float* Bp = B + sbase * DN + tid;
    const float* Cp = C + sbase * DN + tid;
    __builtin_prefetch(Bp, 0, 0);   // global_prefetch_b8
    __builtin_prefetch(Cp, 0, 0);
    __syncthreads();

    const float An = A[tid];
    float h = 0.f, sd = 0.f;
#pragma unroll 4
    for (int t = 0; t < CH; ++t) {
        float dts = dt_s[t];
        sd += dts;
        float a = __expf(dts * An);
        h = fmaf(a, h, dtx_s[t] * Bp[(size_t)t * DN]);
        float p = Cp[(size_t)t * DN] * h;
        // wave32 reduction
        p += __shfl_xor(p, 1, 32);
        p += __shfl_xor(p, 2, 32);
        p += __shfl_xor(p, 4, 32);
        p += __shfl_xor(p, 8, 32);
        p += __shfl_xor(p, 16, 32);
        if (lane == 0) ypart[t * 4 + wv] = p;
    }

    const size_t cidx = ((size_t)b * NCH + k) * DN + tid;
    aprod[cidx] = __expf(An * sd);   // product of all per-step decay factors
    hloc[cidx]  = h;
    __syncthreads();
    y[sbase + tid] = ypart[tid * 4 + 0] + ypart[tid * 4 + 1] +
                     ypart[tid * 4 + 2] + ypart[tid * 4 + 3];
}

// ---------------------------------------------------------------------------
// Pass 2: sequential combine over the 64 chunk summaries (tiny).
// carry[b][k][n] = state entering chunk k; h_final written to d_out tail.
// ---------------------------------------------------------------------------
__global__ void ssm_pass2(const float* __restrict__ aprod, const float* __restrict__ hloc,
                          float* __restrict__ carry, float* __restrict__ hfin)
{
    const int b = blockIdx.x, n = threadIdx.x;
    float h = 0.f;
    for (int k = 0; k < NCH; ++k) {
        const size_t idx = ((size_t)b * NCH + k) * DN + n;
        carry[idx] = h;
        h = fmaf(aprod[idx], h, hloc[idx]);
    }
    hfin[(size_t)b * DN + n] = h;
}

// ---------------------------------------------------------------------------
// Pass 3: cross-chunk correction  y[t] += sum_n C[t,n] * exp(A[n]*Sd[t]) * h_in[n]
// where Sd is the inclusive dt prefix within the chunk. This is an exact f32
// GEMV M[128x128] @ h_in[128], mapped onto V_WMMA_F32_16X16X4_F32:
//   A tile = 16 rows (timesteps) x 4 k of M, B tile = h_in replicated over 16 cols.
// Each of 4 waves owns two 16-row tiles; 32 WMMA issues per tile (K=128/4).
// ---------------------------------------------------------------------------
__global__ void ssm_pass3(const float* __restrict__ dt, const float* __restrict__ C,
                          const float* __restrict__ A, const float* __restrict__ carry,
                          float* __restrict__ y)
{
    __shared__ float Sd[CH];
    __shared__ float sA[DN];
    __shared__ float hin[DN];
    __shared__ float ycorr[CH];

    const int k = blockIdx.x, b = blockIdx.y;
    const int tid = threadIdx.x;
    const int lane = tid & 31, wv = tid >> 5;
    const size_t sbase = (size_t)b * SEQ + (size_t)k * CH;

    Sd[tid]  = dt[sbase + tid];
    sA[tid]  = A[tid];
    hin[tid] = carry[((size_t)b * NCH + k) * DN + tid];
    __syncthreads();
    // inclusive prefix sum of dt over the chunk (Hillis-Steele)
    for (int off = 1; off < CH; off <<= 1) {
        float v = (tid >= off) ? Sd[tid - off] : 0.f;
        __syncthreads();
        Sd[tid] += v;
        __syncthreads();
    }

    const float* Cb = C + sbase * DN;
    const int kBase = (lane < 16) ? 0 : 2;   // K split across lane halves (A/B layout)

    for (int i = 0; i < 2; ++i) {
        const int t0   = (wv * 2 + i) * 16;
        const int tRow = t0 + (lane & 15);
        const float sdt = Sd[tRow];
        const float* Crow = Cb + (size_t)tRow * DN;

        v8f acc = {};
#pragma unroll 8
        for (int k0 = 0; k0 < DN; k0 += 4) {
            const int kk = k0 + kBase;
            float2 cv = *(const float2*)(Crow + kk);
            v2f a2, b2;
            a2.x = cv.x * __expf(sA[kk] * sdt);
            a2.y = cv.y * __expf(sA[kk + 1] * sdt);
            b2.x = hin[kk];
            b2.y = hin[kk + 1];
            acc = __builtin_amdgcn_wmma_f32_16x16x4_f32(
                false, a2, false, b2, (short)0, acc, false, false);
        }
        // D layout: VGPR v, lanes 0-15 -> row v, lanes 16-31 -> row v+8 (all cols equal)
        if (lane == 0) {
#pragma unroll
            for (int r = 0; r < 8; ++r) ycorr[t0 + r] = acc[r];
        }
        if (lane == 16) {
#pragma unroll
            for (int r = 0; r < 8; ++r) ycorr[t0 + 8 + r] = acc[r];
        }
    }
    __syncthreads();
    y[sbase + tid] += ycorr[tid];
}

// ---------------------------------------------------------------------------
extern "C" void kernel_launch(void* const* d_in, const int* in_sizes, int n_in,
                              void* d_out, int out_size, void* d_ws, size_t ws_size,
                              hipStream_t stream) {
    const float* x  = (const float*)d_in[0];   // [8,8192,1]
    const float* dt = (const float*)d_in[1];   // [8,8192]
    const float* B  = (const float*)d_in[2];   // [8,8192,128]
    const float* C  = (const float*)d_in[3];   // [8,8192,128]
    const float* A  = (const float*)d_in[4];   // [128]

    float* y    = (float*)d_out;                       // [8,8192] outputs
    float* hfin = y + (size_t)BATCH * SEQ;             // [8,128] final state

    float* aprod = (float*)d_ws;                       // [8,64,128]
    float* hloc  = aprod + (size_t)BATCH * NCH * DN;   // [8,64,128]
    float* carry = hloc  + (size_t)BATCH * NCH * DN;   // [8,64,128]

    dim3 grid(NCH, BATCH);
    ssm_pass1<<<grid, CH, 0, stream>>>(x, dt, B, C, A, y, aprod, hloc);
    ssm_pass2<<<BATCH, DN, 0, stream>>>(aprod, hloc, carry, hfin);
    ssm_pass3<<<grid, CH, 0, stream>>>(dt, C, A, carry, y);
}